// Mamba2Block_67267777790255
// MI455X (gfx1250) — compile-verified
//
#include <hip/hip_runtime.h>
#include <hip/hip_bf16.h>
#include <math.h>

// ---- Mamba2 dims (fixed by the reference) ----
#define BATCH     2
#define SEQ       2048
#define DMODEL    512
#define DINNER    1024
#define NHEADS    16
#define HEADDIM   64
#define DSTATE    64
#define DCONV     4
#define CONVDIM   1152            // DINNER + 2*DSTATE
#define DINPROJ   2192            // 2*DINNER + 2*DSTATE + NHEADS
#define MROWS     (BATCH*SEQ)     // 4096
#define EPSV      1e-5f

typedef __attribute__((ext_vector_type(2))) float v2f;
typedef __attribute__((ext_vector_type(8))) float v8f;

// ---------------------------------------------------------------------------
// K1/K6: fp32 WMMA GEMM, C[M,N] = A[M,K] * W[N,K]^T  (both row-major).
// 64x64 tile per wave: 4 A-frags x 4 B-frags -> 16 v_wmma per k-step with
// only 8 global_load_b64 (16 FLOP/byte of request traffic, 4x the naive tile).
// Block = 8 waves arranged 2 (M) x 4 (N) -> 128 x 256 block tile.
// A-frag (16x4 f32): lane l holds A[m0+(l&15)][k + 2*(l>>4) + {0,1}]
// B-frag (4x16 f32): lane l holds W[n0+(l&15)][k + 2*(l>>4) + {0,1}]
// D (16x16 f32):     reg r, lane l -> row +r+8*(l>>4), col +(l&15)
// Edge N (GEMM1: N=2192=16*137): clamp B row index (EXEC stays all-ones for
// WMMA), guard stores only.
// ---------------------------------------------------------------------------
__global__ __launch_bounds__(256)
void wmma_gemm_nt(const float* __restrict__ A, const float* __restrict__ W,
                  float* __restrict__ C, int M, int N, int K) {
    const int lane = threadIdx.x & 31;
    const int wave = threadIdx.x >> 5;
    const int wm   = wave & 1;
    const int wn   = wave >> 1;
    const int m0 = blockIdx.y * 128 + wm * 64;
    const int n0 = blockIdx.x * 256 + wn * 64;
    const int mr = lane & 15;
    const int kh = (lane >> 4) << 1;            // 0 or 2: K-half select

    const float* aptr[4];
    const float* bptr[4];
    #pragma unroll
    for (int i = 0; i < 4; ++i) {
        aptr[i] = A + (size_t)(m0 + i * 16 + mr) * K + kh;
        int brow = n0 + i * 16 + mr;
        if (brow > N - 1) brow = N - 1;          // clamp: keep EXEC full
        bptr[i] = W + (size_t)brow * K + kh;
    }

    v8f acc[4][4];
    #pragma unroll
    for (int i = 0; i < 4; ++i)
        #pragma unroll
        for (int j = 0; j < 4; ++j)
            acc[i][j] = (v8f){0.f,0.f,0.f,0.f,0.f,0.f,0.f,0.f};

    for (int k = 0; k < K; k += 4) {
        v2f av[4], bv[4];
        #pragma unroll
        for (int i = 0; i < 4; ++i) av[i] = *(const v2f*)(aptr[i] + k);
        #pragma unroll
        for (int j = 0; j < 4; ++j) bv[j] = *(const v2f*)(bptr[j] + k);
        #pragma unroll
        for (int i = 0; i < 4; ++i)
            #pragma unroll
            for (int j = 0; j < 4; ++j)
                acc[i][j] = __builtin_amdgcn_wmma_f32_16x16x4_f32(
                    false, av[i], false, bv[j], (short)0, acc[i][j], false, false);
    }

    const int rbase = (lane >> 4) << 3;          // 0 or 8
    #pragma unroll
    for (int i = 0; i < 4; ++i) {
        #pragma unroll
        for (int j = 0; j < 4; ++j) {
            const int col = n0 + j * 16 + mr;
            if (col < N) {
                float* cp = C + (size_t)(m0 + i * 16 + rbase) * N + col;
                #pragma unroll
                for (int r = 0; r < 8; ++r) cp[(size_t)r * N] = acc[i][j][r];
            }
        }
    }
}

// ---------------------------------------------------------------------------
// K2: depthwise causal conv (4 taps) + bias + SiLU over the xBC slice
// ---------------------------------------------------------------------------
__global__ __launch_bounds__(256)
void conv_silu_kernel(const float* __restrict__ zxbcdt,
                      const float* __restrict__ conv_w,   // [CONVDIM][DCONV]
                      const float* __restrict__ conv_b,   // [CONVDIM]
                      float* __restrict__ xbc_out) {      // [MROWS][CONVDIM]
    int idx = blockIdx.x * 256 + threadIdx.x;
    if (idx >= MROWS * CONVDIM) return;
    int c   = idx % CONVDIM;
    int row = idx / CONVDIM;
    int b   = row / SEQ;
    int s   = row % SEQ;
    float acc = conv_b[c];
    #pragma unroll
    for (int k = 0; k < DCONV; ++k) {
        int ss = s - (DCONV - 1) + k;
        if (ss >= 0) {
            float xv = zxbcdt[(size_t)(b * SEQ + ss) * DINPROJ + DINNER + c];
            acc = fmaf(xv, conv_w[c * DCONV + k], acc);
        }
    }
    float sig = 1.0f / (1.0f + expf(-acc));
    xbc_out[(size_t)row * CONVDIM + c] = acc * sig;
}

// ---------------------------------------------------------------------------
// K3: dt = softplus(dt_raw + dt_bias)
// ---------------------------------------------------------------------------
__global__ __launch_bounds__(256)
void dt_softplus_kernel(const float* __restrict__ zxbcdt,
                        const float* __restrict__ dt_bias,
                        float* __restrict__ dt_sp) {      // [MROWS][NHEADS]
    int idx = blockIdx.x * 256 + threadIdx.x;
    if (idx >= MROWS * NHEADS) return;
    int h   = idx % NHEADS;
    int row = idx / NHEADS;
    float v = zxbcdt[(size_t)row * DINPROJ + (DINNER + CONVDIM) + h] + dt_bias[h];
    dt_sp[idx] = (v > 20.0f) ? v : log1pf(expf(v));
}

// ---------------------------------------------------------------------------
// K4: selective-state scan. One block per (batch, head); 256 threads:
//   p = tid & 63 (head-dim lane), ng = tid >> 6 (16-state slice of DSTATE).
// State h[p][n] lives in registers (16 per thread). x/B/C staged via LDS,
// y reduced 4-way through LDS.  y += D[h]*x added here.
// ---------------------------------------------------------------------------
__global__ __launch_bounds__(256)
void ssm_scan_kernel(const float* __restrict__ xbc,       // [MROWS][CONVDIM]
                     const float* __restrict__ dt_sp,     // [MROWS][NHEADS]
                     const float* __restrict__ A_log,     // [NHEADS]
                     const float* __restrict__ Dvec,      // [NHEADS]
                     float* __restrict__ y) {             // [MROWS][DINNER]
    const int bh = blockIdx.x;
    const int b  = bh / NHEADS;
    const int h  = bh % NHEADS;
    const int tid = threadIdx.x;
    const int p  = tid & 63;
    const int ng = tid >> 6;

    const float a_h = -expf(A_log[h]);
    const float d_h = Dvec[h];

    __shared__ float sx[HEADDIM];
    __shared__ float sB[DSTATE];
    __shared__ float sC[DSTATE];
    __shared__ float ypart[256];

    float hreg[16];
    #pragma unroll
    for (int i = 0; i < 16; ++i) hreg[i] = 0.0f;

    for (int s = 0; s < SEQ; ++s) {
        const size_t row  = (size_t)(b * SEQ + s);
        const size_t base = row * CONVDIM;
        if (tid < 64)        sx[tid]        = xbc[base + h * HEADDIM + tid];
        else if (tid < 128)  sB[tid - 64]   = xbc[base + DINNER + (tid - 64)];
        else if (tid < 192)  sC[tid - 128]  = xbc[base + DINNER + DSTATE + (tid - 128)];
        const float dtv = dt_sp[row * NHEADS + h];
        const float dA  = __expf(a_h * dtv);
        __syncthreads();

        const float xp   = sx[p];
        const float coef = dtv * xp;
        float ys = 0.0f;
        #pragma unroll
        for (int i = 0; i < 16; ++i) {
            const int n = ng * 16 + i;
            hreg[i] = fmaf(hreg[i], dA, coef * sB[n]);
            ys = fmaf(hreg[i], sC[n], ys);
        }
        ypart[tid] = ys;
        __syncthreads();

        if (ng == 0) {
            float yt = ypart[p] + ypart[64 + p] + ypart[128 + p] + ypart[192 + p];
            y[row * DINNER + h * HEADDIM + p] = fmaf(d_h, xp, yt);
        }
    }
}

// ---------------------------------------------------------------------------
// K5: y = y * silu(z); RMS-norm over DINNER; scale by norm_w.  One block/row.
// ---------------------------------------------------------------------------
__global__ __launch_bounds__(256)
void gate_rmsnorm_kernel(const float* __restrict__ zxbcdt,
                         const float* __restrict__ norm_w,
                         float* __restrict__ y) {          // in/out [MROWS][DINNER]
    const int row = blockIdx.x;
    const int tid = threadIdx.x;
    __shared__ float red[256];

    float ss = 0.0f;
    #pragma unroll
    for (int j = tid; j < DINNER; j += 256) {
        float yv = y[(size_t)row * DINNER + j];
        float zv = zxbcdt[(size_t)row * DINPROJ + j];
        float g  = yv * (zv / (1.0f + expf(-zv)));
        y[(size_t)row * DINNER + j] = g;
        ss = fmaf(g, g, ss);
    }
    red[tid] = ss;
    __syncthreads();
    for (int s = 128; s > 0; s >>= 1) {
        if (tid < s) red[tid] += red[tid + s];
        __syncthreads();
    }
    const float scale = rsqrtf(red[0] * (1.0f / DINNER) + EPSV);
    #pragma unroll
    for (int j = tid; j < DINNER; j += 256) {
        y[(size_t)row * DINNER + j] = y[(size_t)row * DINNER + j] * scale * norm_w[j];
    }
}

// ---------------------------------------------------------------------------
extern "C" void kernel_launch(void* const* d_in, const int* in_sizes, int n_in,
                              void* d_out, int out_size, void* d_ws, size_t ws_size,
                              hipStream_t stream) {
    const float* u       = (const float*)d_in[0];
    const float* W_in    = (const float*)d_in[1];
    const float* conv_w  = (const float*)d_in[2];
    const float* conv_b  = (const float*)d_in[3];
    const float* dt_bias = (const float*)d_in[4];
    const float* A_log   = (const float*)d_in[5];
    const float* Dvec    = (const float*)d_in[6];
    const float* norm_w  = (const float*)d_in[7];
    const float* W_out   = (const float*)d_in[8];
    float* out = (float*)d_out;

    // Workspace layout (floats)
    float* ws      = (float*)d_ws;
    float* zxbcdt  = ws;                                   // 4096*2192
    float* xbc     = zxbcdt + (size_t)MROWS * DINPROJ;     // 4096*1152
    float* dt_sp   = xbc    + (size_t)MROWS * CONVDIM;     // 4096*16
    float* ybuf    = dt_sp  + (size_t)MROWS * NHEADS;      // 4096*1024

    // K1: zxbcdt = u @ W_in^T    (M=4096, N=2192, K=512)
    {
        dim3 grid((DINPROJ + 255) / 256, MROWS / 128);
        wmma_gemm_nt<<<grid, 256, 0, stream>>>(u, W_in, zxbcdt, MROWS, DINPROJ, DMODEL);
    }
    // K2: depthwise conv + SiLU
    {
        int total = MROWS * CONVDIM;
        conv_silu_kernel<<<(total + 255) / 256, 256, 0, stream>>>(zxbcdt, conv_w, conv_b, xbc);
    }
    // K3: dt softplus
    {
        int total = MROWS * NHEADS;
        dt_softplus_kernel<<<(total + 255) / 256, 256, 0, stream>>>(zxbcdt, dt_bias, dt_sp);
    }
    // K4: SSM scan (32 independent chains)
    ssm_scan_kernel<<<BATCH * NHEADS, 256, 0, stream>>>(xbc, dt_sp, A_log, Dvec, ybuf);

    // K5: gate + RMS norm (in place on ybuf)
    gate_rmsnorm_kernel<<<MROWS, 256, 0, stream>>>(zxbcdt, norm_w, ybuf);

    // K6: out = ybuf @ W_out^T   (M=4096, N=512, K=1024)
    {
        dim3 grid(DMODEL / 256, MROWS / 128);
        wmma_gemm_nt<<<grid, 256, 0, stream>>>(ybuf, W_out, out, MROWS, DMODEL, DINNER);
    }
}